// Deconv2d_83854941487369
// MI455X (gfx1250) — compile-verified
//
#include <hip/hip_runtime.h>

#if defined(__HIP_DEVICE_COMPILE__)
#if !__has_builtin(__builtin_amdgcn_wmma_f32_16x16x4_f32)
#error "device pass: missing __builtin_amdgcn_wmma_f32_16x16x4_f32"
#endif
#endif

typedef __attribute__((ext_vector_type(2))) float v2f;
typedef __attribute__((ext_vector_type(8))) float v8f;

// Problem constants (from reference)
#define BN    8      // batch
#define CI    256    // input channels
#define CO    128    // output channels
#define HH    64     // x height
#define WW    64     // x width
#define OH    130    // out height = (64-1)*2+4
#define OW    130    // out width
#define UU    65     // oy = p + 2u, u in [0,65)
#define NPB   (UU*UU)                   // 4225 pixels per (b, parity)
#define BLK_N 128                       // pixels per block
#define NT    ((NPB + BLK_N - 1)/BLK_N) // 34 N-tiles

// Block = 256 threads = 8 waves laid out 2(M) x 4(N).
// Each wave computes a 32(M) x 32(N) tile = 2x2 WMMA subtiles with 4
// f32 accumulators; K = 4 taps * 256 channels = 1024, stepped 4 wide
// through v_wmma_f32_16x16x4_f32. A fragments are shared across the two
// N-subtiles and B fragments across the two M-subtiles -> 2 VMEM dwords
// per WMMA per lane, and a RAW distance of 4 between same-acc WMMAs.
__global__ __launch_bounds__(256)
void deconv2d_fold_wmma(const float* __restrict__ x,
                        const float* __restrict__ wgt,
                        const float* __restrict__ bias,
                        float* __restrict__ out)
{
    const int tile = blockIdx.x;         // N tile (128 pixels)
    const int bb   = blockIdx.y;         // batch
    const int mt   = blockIdx.z >> 2;    // M tile (64 rows)
    const int par  = blockIdx.z & 3;     // parity pair
    const int p = par >> 1, q = par & 1;

    const int wave = threadIdx.x >> 5;
    const int lane = threadIdx.x & 31;
    const int l16  = lane & 15;
    const int hi   = lane >> 4;          // selects K pair (A/B) and M+8 (D)
    const int kk   = hi * 2;

    const int wr = wave >> 2;            // 0..1 : M sub-block (32 rows)
    const int wcn = wave & 3;            // 0..3 : N sub-block (32 cols)
    const int mw = mt * 64 + wr * 32;    // wave M base
    const int nw = tile * BLK_N + wcn * 32; // wave N base (pixels)

    // Per N-subtile (sj) and tap: x pointer (clamped in-bounds) + 0/1 mask.
    const float* xp[2][4];
    float        mk[2][4];
    bool         nval[2];
    int          uu[2], vv[2];
    #pragma unroll
    for (int sj = 0; sj < 2; ++sj) {
        const int n = nw + sj * 16 + l16;
        nval[sj] = (n < NPB);
        const int nc = nval[sj] ? n : (NPB - 1);
        uu[sj] = nc / UU;
        vv[sj] = nc % UU;
        #pragma unroll
        for (int t = 0; t < 4; ++t) {
            const int dy = t & 1, dx = t >> 1;
            const int h = uu[sj] - dy;
            const int w = vv[sj] - dx;
            const bool ok = (h >= 0) && (h < HH) && (w >= 0) && (w < WW);
            const int hc = h < 0 ? 0 : (h > HH - 1 ? HH - 1 : h);
            const int wc2 = w < 0 ? 0 : (w > WW - 1 ? WW - 1 : w);
            xp[sj][t] = x + (size_t)bb * CI * HH * WW
                          + (size_t)kk * HH * WW
                          + (size_t)hc * WW + wc2;
            mk[sj][t] = ok ? 1.0f : 0.0f;
        }
    }

    // Per M-subtile (si) and tap: weight pointer (A row = l16).
    const float* wp[2][4];
    #pragma unroll
    for (int si = 0; si < 2; ++si) {
        #pragma unroll
        for (int t = 0; t < 4; ++t) {
            const int dy = t & 1, dx = t >> 1;
            const int ky = p + 2 * dy, kx = q + 2 * dx;
            wp[si][t] = wgt + (size_t)(mw + si * 16 + l16) * (CI * 16)
                            + (size_t)kk * 16
                            + (size_t)ky * 4 + kx;
        }
    }

    v8f acc[2][2] = {};

    #pragma unroll
    for (int t = 0; t < 4; ++t) {
        const float* __restrict__ wq0 = wp[0][t];
        const float* __restrict__ wq1 = wp[1][t];
        const float* __restrict__ xq0 = xp[0][t];
        const float* __restrict__ xq1 = xp[1][t];
        const float m0 = mk[0][t], m1 = mk[1][t];
        #pragma unroll 4
        for (int i0 = 0; i0 < CI; i0 += 4) {
            v2f a0, a1, b0, b1;
            a0.x = wq0[(size_t)i0 * 16];
            a0.y = wq0[(size_t)i0 * 16 + 16];
            a1.x = wq1[(size_t)i0 * 16];
            a1.y = wq1[(size_t)i0 * 16 + 16];
            b0.x = xq0[(size_t)i0 * (HH * WW)] * m0;
            b0.y = xq0[(size_t)i0 * (HH * WW) + HH * WW] * m0;
            b1.x = xq1[(size_t)i0 * (HH * WW)] * m1;
            b1.y = xq1[(size_t)i0 * (HH * WW) + HH * WW] * m1;
            acc[0][0] = __builtin_amdgcn_wmma_f32_16x16x4_f32(
                false, a0, false, b0, (short)0, acc[0][0], false, false);
            acc[0][1] = __builtin_amdgcn_wmma_f32_16x16x4_f32(
                false, a0, false, b1, (short)0, acc[0][1], false, false);
            acc[1][0] = __builtin_amdgcn_wmma_f32_16x16x4_f32(
                false, a1, false, b0, (short)0, acc[1][0], false, false);
            acc[1][1] = __builtin_amdgcn_wmma_f32_16x16x4_f32(
                false, a1, false, b1, (short)0, acc[1][1], false, false);
        }
    }

    // Epilogue: D layout -> VGPR j holds (M = j + 8*hi, N = l16).
    #pragma unroll
    for (int sj = 0; sj < 2; ++sj) {
        if (!nval[sj]) continue;
        const int oy = p + 2 * uu[sj];
        const int ox = q + 2 * vv[sj];
        const size_t ob = (size_t)bb * CO * OH * OW + (size_t)oy * OW + ox;
        #pragma unroll
        for (int si = 0; si < 2; ++si) {
            #pragma unroll
            for (int j = 0; j < 8; ++j) {
                const int m = mw + si * 16 + j + hi * 8;
                float bs = 0.0f;
                #pragma unroll
                for (int t = 0; t < 4; ++t) {
                    const int dy = t & 1, dx = t >> 1;
                    bs += mk[sj][t] * bias[m * 16 + (p + 2 * dy) * 4 + (q + 2 * dx)];
                }
                out[ob + (size_t)m * OH * OW] = acc[si][sj][j] + bs;
            }
        }
    }
}

extern "C" void kernel_launch(void* const* d_in, const int* in_sizes, int n_in,
                              void* d_out, int out_size, void* d_ws, size_t ws_size,
                              hipStream_t stream) {
    (void)in_sizes; (void)n_in; (void)out_size; (void)d_ws; (void)ws_size;
    const float* x    = (const float*)d_in[0];   // (8,256,64,64)
    const float* wgt  = (const float*)d_in[1];   // (128,256,4,4)
    const float* bias = (const float*)d_in[2];   // (128,4,4)
    float*       out  = (float*)d_out;           // (8,128,130,130)

    dim3 grid(NT, BN, 8);   // 34 x 8 x (2 M-tiles * 4 parities)
    dim3 block(256);        // 8 waves, wave32
    deconv2d_fold_wmma<<<grid, block, 0, stream>>>(x, wgt, bias, out);
}